// LocalTokenDispatcher_22874995818747
// MI455X (gfx1250) — compile-verified
//
#include <hip/hip_runtime.h>
#include <hip/hip_bf16.h>

// MoE local-token-dispatcher with identity experts collapses to:
//   out[t,d]  = x[t,d]*s[t,0] + x[t,d]*s[t,1]   (pure streaming, HBM-bound)
//   counts[e] = histogram of selected_experts_indices
//
// MI455X roofline: 512 MB of traffic @ 23.3 TB/s => ~22 us. Compute is
// negligible, so the kernel is built around b128 non-temporal loads/stores
// and block-uniform score loads.

#define NUM_EXPERTS 64
#define TOP_K       2
#define NUM_TOKENS  16384
#define DIM         4096

typedef float v4f __attribute__((ext_vector_type(4)));

// One workgroup per token row: 256 threads x 4 float4 = 1024 float4 = 4096 floats.
__global__ __launch_bounds__(256) void moe_scale_rows_kernel(
    const float* __restrict__ x,
    const float* __restrict__ top_scores,
    float* __restrict__ out)
{
    const int t   = blockIdx.x;          // token id, uniform per block
    const int tid = threadIdx.x;

    // Block-uniform score pair: 8 bytes per 16 KB row (scalar-loadable).
    const float s0 = top_scores[2 * t + 0];
    const float s1 = top_scores[2 * t + 1];

    const v4f* __restrict__ xrow = (const v4f*)(x + (size_t)t * DIM);
    v4f* __restrict__ orow       = (v4f*)(out + (size_t)t * DIM);

    // Warm GL2 a few rows ahead: 256 lanes * 64B spacing covers one 16 KB row.
    // gfx1250 global_prefetch_b8; RT-speculative, no counter cost.
    if (t + 4 < NUM_TOKENS) {
        __builtin_prefetch(x + (size_t)(t + 4) * DIM + tid * 16, 0, 1);
    }

#pragma unroll
    for (int j = 0; j < 4; ++j) {
        const int idx = j * 256 + tid;           // coalesced b128 across the wave
        v4f v = __builtin_nontemporal_load(&xrow[idx]);   // read-once stream: TH=NT
        v4f o;
        {
            // Match reference rounding: round each product, then add
            // (no fma contraction across the two products).
#pragma clang fp contract(off)
            o.x = v.x * s0 + v.x * s1;
            o.y = v.y * s0 + v.y * s1;
            o.z = v.z * s0 + v.z * s1;
            o.w = v.w * s0 + v.w * s1;
        }
        __builtin_nontemporal_store(o, &orow[idx]);       // write-once stream: TH=NT
    }
}

// 32768 expert ids -> 64 float bins. Single 1024-thread workgroup with an LDS
// histogram (ds_add_u32); final bins stored directly (no global zero-init pass
// needed, deterministic across replays).
__global__ __launch_bounds__(1024) void moe_counts_kernel(
    const int* __restrict__ experts,
    float* __restrict__ counts)
{
    __shared__ int bins[NUM_EXPERTS];
    const int tid = threadIdx.x;

    if (tid < NUM_EXPERTS) bins[tid] = 0;
    __syncthreads();

    const int total = NUM_TOKENS * TOP_K;   // 32768
#pragma unroll 4
    for (int i = tid; i < total; i += 1024) {
        atomicAdd(&bins[experts[i] & (NUM_EXPERTS - 1)], 1);
    }
    __syncthreads();

    if (tid < NUM_EXPERTS) counts[tid] = (float)bins[tid];
}

extern "C" void kernel_launch(void* const* d_in, const int* in_sizes, int n_in,
                              void* d_out, int out_size, void* d_ws, size_t ws_size,
                              hipStream_t stream)
{
    (void)in_sizes; (void)n_in; (void)out_size; (void)d_ws; (void)ws_size;

    const float* x       = (const float*)d_in[0];   // (16384, 4096) f32
    const float* scores  = (const float*)d_in[1];   // (16384, 2)    f32
    const int*   experts = (const int*)d_in[2];     // (16384, 2)    i32
    // d_in[3] (num_tokens_per_expert) is an all-zero input; unused.

    float* out    = (float*)d_out;                          // (16384, 4096) f32
    float* counts = (float*)d_out + (size_t)NUM_TOKENS * DIM; // (64,) f32

    moe_scale_rows_kernel<<<NUM_TOKENS, 256, 0, stream>>>(x, scores, out);
    moe_counts_kernel<<<1, 1024, 0, stream>>>(experts, counts);
}